// loss_kernelMatching_perpixel_ms_14499809591889
// MI455X (gfx1250) — compile-verified
//
#include <hip/hip_runtime.h>

// Problem constants (fixed by the reference).
#define PP     1681          // pixels per row (41*41)
#define BS     8             // batch
#define NT     4             // number of scales
#define NROWS  (BS * PP)     // 13448 combined (b, j) rows
#define NTILE  106           // ceil(1681 / 16) row tiles per batch
#define NCHUNK 421           // ceil(1681 / 4)  K-chunks of 4 columns
#define WPB    8             // waves per block in the main kernel
#define LDSW   (PP + 3)      // padded w-row length (j0+1 can reach 1683)

typedef __attribute__((ext_vector_type(2))) float v2f;
typedef __attribute__((ext_vector_type(8))) float v8f;

__device__ __forceinline__ float wave_reduce_add(float x) {
  #pragma unroll
  for (int off = 16; off > 0; off >>= 1) x += __shfl_xor(x, off, 32);
  return x;
}

// ---------------------------------------------------------------------------
// Kernel 1: inv_gt[b*P+j] = rsqrt( sum_k target[b,j,k]^2 )
// one wave per row; coalesced 128B strides per wave.
// ---------------------------------------------------------------------------
__global__ __launch_bounds__(256) void k_tgt_norm(const float* __restrict__ tgt,
                                                  float* __restrict__ inv_gt) {
  const int wave = __builtin_amdgcn_readfirstlane(threadIdx.x >> 5);
  const int lane = threadIdx.x & 31;
  const int r = blockIdx.x * 8 + wave;           // r in [0, NROWS); grid = 1681
  const float* row = tgt + (size_t)r * PP;
  float s = 0.f;
  for (int k = lane; k < PP; k += 32) { float v = row[k]; s = fmaf(v, v, s); }
  s = wave_reduce_add(s);
  if (lane == 0) inv_gt[r] = __builtin_amdgcn_rsqf(s);
}

// ---------------------------------------------------------------------------
// Kernel 2: w[t][r] = rsqrt( sum_k pred[r,k]^(2*ts_t) ) * inv_gt[r]
// pred^x = exp2(x * log2(pred)); one log2, four exp2 per element.
// ---------------------------------------------------------------------------
__global__ __launch_bounds__(256) void k_pred_norm(const float* __restrict__ pred,
                                                   const float* __restrict__ ts4,
                                                   const float* __restrict__ inv_gt,
                                                   float* __restrict__ wpl) {
  const int wave = __builtin_amdgcn_readfirstlane(threadIdx.x >> 5);
  const int lane = threadIdx.x & 31;
  const int r = blockIdx.x * 8 + wave;
  const float t0 = 2.f * ts4[0], t1 = 2.f * ts4[1], t2 = 2.f * ts4[2], t3 = 2.f * ts4[3];
  const float* row = pred + (size_t)r * PP;
  float s0 = 0.f, s1 = 0.f, s2 = 0.f, s3 = 0.f;
  for (int k = lane; k < PP; k += 32) {
    float l = __builtin_amdgcn_logf(row[k]);     // v_log_f32 (base 2)
    s0 += __builtin_amdgcn_exp2f(t0 * l);
    s1 += __builtin_amdgcn_exp2f(t1 * l);
    s2 += __builtin_amdgcn_exp2f(t2 * l);
    s3 += __builtin_amdgcn_exp2f(t3 * l);
  }
  s0 = wave_reduce_add(s0); s1 = wave_reduce_add(s1);
  s2 = wave_reduce_add(s2); s3 = wave_reduce_add(s3);
  if (lane == 0) {
    float ig = inv_gt[r];
    wpl[0 * NROWS + r] = __builtin_amdgcn_rsqf(s0) * ig;
    wpl[1 * NROWS + r] = __builtin_amdgcn_rsqf(s1) * ig;
    wpl[2 * NROWS + r] = __builtin_amdgcn_rsqf(s2) * ig;
    wpl[3 * NROWS + r] = __builtin_amdgcn_rsqf(s3) * ig;
  }
}

// ---------------------------------------------------------------------------
// Kernel 3: main loop. One block = one 16-row tile of one batch. 8 waves split
// the j range in chunks of 4. Per chunk, per scale t:
//   A(16x4)[i][k] = pred[i,jb+k]^ts_t * target[i,jb+k]   (lane-local compute)
//   B(4x16)[k][n] = w_t[jb+k]                             (broadcast from LDS)
//   C_t += A x B  via V_WMMA_F32_16X16X4_F32  -> every C column = partial s_t
// A-layout (ISA 7.12.2): lane L<16 holds row L, K={0,1}; lane L>=16 row L-16,
// K={2,3}; B mirrors it, so both use j0 = jb + 2*(lane>>4).
// Hot loop covers chunks 0..NCHUNK-2 (always in-range, no guards); the single
// ragged tail chunk is done by its owning wave under a scalar branch.
// ---------------------------------------------------------------------------
__global__ __launch_bounds__(256) void k_main(const float* __restrict__ pred,
                                              const float* __restrict__ tgt,
                                              const float* __restrict__ ts4,
                                              const float* __restrict__ wpl,
                                              float* __restrict__ partials) {
  __shared__ float lds_w[NT][LDSW];
  __shared__ float lds_part[WPB][NT][16];
  __shared__ float lds_row[16];

  const int b = blockIdx.x / NTILE;
  const int tile = blockIdx.x % NTILE;
  const int rowbase = tile * 16;
  const int tid = threadIdx.x;
  const int wave = __builtin_amdgcn_readfirstlane(tid >> 5);
  const int lane = tid & 31;

  // stage the 4 weight vectors for this batch into LDS (zero-padded tail)
  for (int idx = tid; idx < NT * LDSW; idx += 256) {
    int t = idx / LDSW, j = idx % LDSW;
    lds_w[t][j] = (j < PP) ? wpl[t * NROWS + b * PP + j] : 0.f;
  }
  __syncthreads();

  const float ts0 = ts4[0], ts1 = ts4[1], ts2 = ts4[2], ts3 = ts4[3];

  const int row = rowbase + (lane & 15);
  const bool row_ok = row < PP;
  const int rclamp = row_ok ? row : (PP - 1);
  const float rmask = row_ok ? 1.f : 0.f;        // zero out padded rows
  const float* prow = pred + ((size_t)b * PP + rclamp) * PP;
  const float* trow = tgt  + ((size_t)b * PP + rclamp) * PP;
  const int joff = (lane >> 4) << 1;             // 0 for lanes 0-15, 2 for 16-31

  v8f c0 = {}, c1 = {}, c2 = {}, c3 = {};

  // -------- guard-free hot loop: chunks 0 .. NCHUNK-2 --------
  #pragma unroll 2
  for (int c = wave; c < NCHUNK - 1; c += WPB) {
    const int j0 = c * 4 + joff;
    const float p0 = prow[j0];
    const float p1 = prow[j0 + 1];
    const float g0 = trow[j0] * rmask;
    const float g1 = trow[j0 + 1] * rmask;
    const float l0 = __builtin_amdgcn_logf(p0);
    const float l1 = __builtin_amdgcn_logf(p1);

    v2f A, B;
    A.x = __builtin_amdgcn_exp2f(ts0 * l0) * g0;
    A.y = __builtin_amdgcn_exp2f(ts0 * l1) * g1;
    B.x = lds_w[0][j0]; B.y = lds_w[0][j0 + 1];
    c0 = __builtin_amdgcn_wmma_f32_16x16x4_f32(false, A, false, B, (short)0, c0, false, false);

    A.x = __builtin_amdgcn_exp2f(ts1 * l0) * g0;
    A.y = __builtin_amdgcn_exp2f(ts1 * l1) * g1;
    B.x = lds_w[1][j0]; B.y = lds_w[1][j0 + 1];
    c1 = __builtin_amdgcn_wmma_f32_16x16x4_f32(false, A, false, B, (short)0, c1, false, false);

    A.x = __builtin_amdgcn_exp2f(ts2 * l0) * g0;
    A.y = __builtin_amdgcn_exp2f(ts2 * l1) * g1;
    B.x = lds_w[2][j0]; B.y = lds_w[2][j0 + 1];
    c2 = __builtin_amdgcn_wmma_f32_16x16x4_f32(false, A, false, B, (short)0, c2, false, false);

    A.x = __builtin_amdgcn_exp2f(ts3 * l0) * g0;
    A.y = __builtin_amdgcn_exp2f(ts3 * l1) * g1;
    B.x = lds_w[3][j0]; B.y = lds_w[3][j0 + 1];
    c3 = __builtin_amdgcn_wmma_f32_16x16x4_f32(false, A, false, B, (short)0, c3, false, false);
  }

  // -------- ragged tail chunk (jb = 1680): owned by one wave --------
  if (wave == (NCHUNK - 1) % WPB) {
    const int j0 = (NCHUNK - 1) * 4 + joff;      // 1680 or 1682
    const int j1 = j0 + 1;
    const int jj0 = (j0 < PP) ? j0 : (PP - 1);
    const int jj1 = (j1 < PP) ? j1 : (PP - 1);
    const float m0 = (j0 < PP) ? rmask : 0.f;
    const float m1 = (j1 < PP) ? rmask : 0.f;
    const float p0 = prow[jj0];
    const float p1 = prow[jj1];
    const float g0 = trow[jj0] * m0;
    const float g1 = trow[jj1] * m1;
    const float l0 = __builtin_amdgcn_logf(p0);
    const float l1 = __builtin_amdgcn_logf(p1);

    v2f A, B;
    A.x = __builtin_amdgcn_exp2f(ts0 * l0) * g0;
    A.y = __builtin_amdgcn_exp2f(ts0 * l1) * g1;
    B.x = lds_w[0][j0]; B.y = lds_w[0][j1];      // LDS padded with zeros past PP
    c0 = __builtin_amdgcn_wmma_f32_16x16x4_f32(false, A, false, B, (short)0, c0, false, false);

    A.x = __builtin_amdgcn_exp2f(ts1 * l0) * g0;
    A.y = __builtin_amdgcn_exp2f(ts1 * l1) * g1;
    B.x = lds_w[1][j0]; B.y = lds_w[1][j1];
    c1 = __builtin_amdgcn_wmma_f32_16x16x4_f32(false, A, false, B, (short)0, c1, false, false);

    A.x = __builtin_amdgcn_exp2f(ts2 * l0) * g0;
    A.y = __builtin_amdgcn_exp2f(ts2 * l1) * g1;
    B.x = lds_w[2][j0]; B.y = lds_w[2][j1];
    c2 = __builtin_amdgcn_wmma_f32_16x16x4_f32(false, A, false, B, (short)0, c2, false, false);

    A.x = __builtin_amdgcn_exp2f(ts3 * l0) * g0;
    A.y = __builtin_amdgcn_exp2f(ts3 * l1) * g1;
    B.x = lds_w[3][j0]; B.y = lds_w[3][j1];
    c3 = __builtin_amdgcn_wmma_f32_16x16x4_f32(false, A, false, B, (short)0, c3, false, false);
  }

  // C columns are all identical: VGPR v / lanes 0-15 = row v, lanes 16-31 = row 8+v
  if (lane == 0 || lane == 16) {
    const int half = lane >> 4;
    #pragma unroll
    for (int v = 0; v < 8; ++v) {
      lds_part[wave][0][half * 8 + v] = c0[v];
      lds_part[wave][1][half * 8 + v] = c1[v];
      lds_part[wave][2][half * 8 + v] = c2[v];
      lds_part[wave][3][half * 8 + v] = c3[v];
    }
  }
  __syncthreads();

  if (tid < 16) {
    float s0 = 0.f, s1 = 0.f, s2 = 0.f, s3 = 0.f;
    #pragma unroll
    for (int w = 0; w < WPB; ++w) {
      s0 += lds_part[w][0][tid];
      s1 += lds_part[w][1][tid];
      s2 += lds_part[w][2][tid];
      s3 += lds_part[w][3][tid];
    }
    // min_t(-s_t) = -max_t(s_t); padded rows give s==0 -> contribute 0
    lds_row[tid] = -fmaxf(fmaxf(s0, s1), fmaxf(s2, s3));
  }
  __syncthreads();
  if (tid == 0) {
    float tot = 0.f;
    #pragma unroll
    for (int i = 0; i < 16; ++i) tot += lds_row[i];
    partials[blockIdx.x] = tot;
  }
}

// ---------------------------------------------------------------------------
// Kernel 4: deterministic final reduction + scaling.
// ---------------------------------------------------------------------------
__global__ __launch_bounds__(256) void k_final(const float* __restrict__ partials,
                                               int n, float* __restrict__ out) {
  __shared__ float red[256];
  const int tid = threadIdx.x;
  float s = 0.f;
  for (int i = tid; i < n; i += 256) s += partials[i];
  red[tid] = s;
  __syncthreads();
  for (int off = 128; off > 0; off >>= 1) {
    if (tid < off) red[tid] += red[tid + off];
    __syncthreads();
  }
  if (tid == 0) out[0] = red[0] / (float)(BS * PP);
}

extern "C" void kernel_launch(void* const* d_in, const int* in_sizes, int n_in,
                              void* d_out, int out_size, void* d_ws, size_t ws_size,
                              hipStream_t stream) {
  const float* pred = (const float*)d_in[0];
  const float* tgt  = (const float*)d_in[1];
  const float* ts4  = (const float*)d_in[2];
  float* out = (float*)d_out;

  float* ws       = (float*)d_ws;
  float* inv_gt   = ws;                 // NROWS floats
  float* wpl      = ws + NROWS;         // 4*NROWS floats (t-major planes)
  float* partials = ws + 5 * NROWS;     // BS*NTILE = 848 floats

  k_tgt_norm <<<NROWS / 8, 256, 0, stream>>>(tgt, inv_gt);
  k_pred_norm<<<NROWS / 8, 256, 0, stream>>>(pred, ts4, inv_gt, wpl);
  k_main     <<<BS * NTILE, 256, 0, stream>>>(pred, tgt, ts4, wpl, partials);
  k_final    <<<1, 256, 0, stream>>>(partials, BS * NTILE, out);
}